// PointerNetwork_65335042507133
// MI455X (gfx1250) — compile-verified
//
#include <hip/hip_runtime.h>
#include <hip/hip_bf16.h>

// Problem constants (from reference setup_inputs): B=4, Ne=Nd=512, C=256
#define BB 4
#define NE 512
#define ND 512
#define CC 256

typedef __attribute__((ext_vector_type(2))) float v2f;
typedef __attribute__((ext_vector_type(8))) float v8f;

__device__ __forceinline__ float fast_tanh(float x) {
#if __has_builtin(__builtin_amdgcn_tanhf)
    return __builtin_amdgcn_tanhf(x);
#else
    float r;
    asm volatile("v_tanh_f32 %0, %1\n\tv_nop" : "=v"(r) : "v"(x));
    return r;
#endif
}

// ---------------------------------------------------------------------------
// Stage A: Y[m][n] = sum_k X[m][k] * W[n][k]   (Y = X @ W^T)
// X: (2048, 256) row-major, W: (256, 256) row-major, Y: (2048, 256) row-major
// One wave per 16x16 output tile, K-loop of V_WMMA_F32_16X16X4_F32.
// Fragment layouts per CDNA5 ISA 7.12.2:
//   A (16x4):  lane r=lane&15 -> row M=r;  VGPR j, half=lane>>4 -> K = 2*half + j
//   B (4x16):  lane r -> col N=r;          VGPR j, half        -> K = 2*half + j
//   D (16x16): VGPR i -> row M = i + 8*half; col N = r
// ---------------------------------------------------------------------------
__global__ void __launch_bounds__(32)
proj_wmma_f32(const float* __restrict__ X, const float* __restrict__ W,
              float* __restrict__ Y) {
    const int mt   = blockIdx.x;          // 0..127  (M tiles over 2048 rows)
    const int nt   = blockIdx.y;          // 0..15   (N tiles over 256 cols)
    const int lane = threadIdx.x;
    const int half = lane >> 4;
    const int r    = lane & 15;

    const float* ap = X + (size_t)(mt * 16 + r) * CC + 2 * half;
    const float* bp = W + (size_t)(nt * 16 + r) * CC + 2 * half;

    v8f acc = {};
#pragma unroll 8
    for (int k = 0; k < CC; k += 4) {
        v2f a = *(const v2f*)(ap + k);
        v2f b = *(const v2f*)(bp + k);
        // 8 args: (neg_a, A, neg_b, B, c_mod, C, reuse_a, reuse_b)
        acc = __builtin_amdgcn_wmma_f32_16x16x4_f32(
            false, a, false, b, (short)0, acc, false, false);
    }

    float* yp = Y + (size_t)(mt * 16 + 8 * half) * CC + nt * 16 + r;
#pragma unroll
    for (int i = 0; i < 8; ++i) yp[(size_t)i * CC] = acc[i];
}

// ---------------------------------------------------------------------------
// Stage B: fused tanh-attention + masked softmax.
// Block = (b, 16-e tile), 512 threads = 16 waves; wave w owns e-row etile*16+w.
// Lane l owns d = t*32 + l for each of 16 d-tiles -> 16 logits in registers.
// enc staged in LDS 32 rows/tile with stride 260 floats (2-cycle b128, 16B ok).
// Softmax is a pure intra-wave reduction (wave32).
// ---------------------------------------------------------------------------
#define EPAD 260   // 256 + 4 pad floats; 260 % 64 == 4, rows 16B-aligned

__global__ void __launch_bounds__(512)
attn_tanh_softmax(const float* __restrict__ enc_ws,
                  const float* __restrict__ dec_ws,
                  const int*   __restrict__ mask,
                  const float* __restrict__ v,
                  float*       __restrict__ out) {
    __shared__ float dec_s[16 * CC];
    __shared__ float v_s[CC];
    __shared__ float enc_s[32 * EPAD];

    const int tid   = threadIdx.x;
    const int etile = blockIdx.x;     // 0..31
    const int b     = blockIdx.y;     // 0..3
    const int w     = tid >> 5;       // wave id = e row within tile
    const int lane  = tid & 31;
    const int e     = etile * 16 + w;

    // Load dec rows for this e-tile (16 x 256 floats) and v (256 floats).
    {
        const int r  = tid >> 5;              // 0..15
        const int cb = (tid & 31) * 4;        // 0..124
        const float* src = dec_ws + ((size_t)(b * NE + etile * 16 + r)) * CC;
        *(float4*)&dec_s[r * CC + cb]       = *(const float4*)&src[cb];
        *(float4*)&dec_s[r * CC + cb + 128] = *(const float4*)&src[cb + 128];
    }
    if (tid < 64) ((float4*)v_s)[tid] = ((const float4*)v)[tid];

    float logit[16];

    for (int t = 0; t < 16; ++t) {
        __syncthreads();   // previous tile fully consumed (also fences dec/v at t=0)
        {
            const int r  = tid >> 4;          // 0..31 enc row within tile
            const int cb = (tid & 15) * 4;    // 0..60
            const float* src = enc_ws + ((size_t)(b * ND + t * 32 + r)) * CC;
#pragma unroll
            for (int j = 0; j < 4; ++j)
                *(float4*)&enc_s[r * EPAD + cb + j * 64] =
                    *(const float4*)&src[cb + j * 64];
            if (t < 15)   // global_prefetch_b8 for the next tile
                __builtin_prefetch(
                    enc_ws + ((size_t)(b * ND + (t + 1) * 32 + r)) * CC + cb, 0, 0);
        }
        __syncthreads();

        const float4* dp = (const float4*)&dec_s[w * CC];     // broadcast
        const float4* ep = (const float4*)&enc_s[lane * EPAD];
        const float4* vp = (const float4*)v_s;                // broadcast
        float acc = 0.f;
#pragma unroll 8
        for (int c4 = 0; c4 < CC / 4; ++c4) {
            float4 dv = dp[c4];
            float4 ev = ep[c4];
            float4 vv = vp[c4];
            acc += vv.x * fast_tanh(dv.x + ev.x);
            acc += vv.y * fast_tanh(dv.y + ev.y);
            acc += vv.z * fast_tanh(dv.z + ev.z);
            acc += vv.w * fast_tanh(dv.w + ev.w);
        }
        const int d = t * 32 + lane;
        const int m = mask[((size_t)(b * NE + e)) * ND + d];
        // log(mask + 1e-16): 0 for mask=1, log(1e-16) for mask=0
        logit[t] = acc + (m ? 0.f : -36.8413615f);
    }

    // Wave-local masked softmax over d (16 regs x 32 lanes = 512 values).
    float mx = logit[0];
#pragma unroll
    for (int t = 1; t < 16; ++t) mx = fmaxf(mx, logit[t]);
#pragma unroll
    for (int off = 16; off > 0; off >>= 1)
        mx = fmaxf(mx, __shfl_xor(mx, off, 32));

    float sum = 0.f;
#pragma unroll
    for (int t = 0; t < 16; ++t) {
        float p = __expf(logit[t] - mx);
        logit[t] = p;
        sum += p;
    }
#pragma unroll
    for (int off = 16; off > 0; off >>= 1)
        sum += __shfl_xor(sum, off, 32);

    const float inv = 1.f / sum;
    float* op = out + ((size_t)(b * NE + e)) * ND;
#pragma unroll
    for (int t = 0; t < 16; ++t) op[t * 32 + lane] = logit[t] * inv;
}

// ---------------------------------------------------------------------------
// Launch. Inputs (setup_inputs order): x_decoder, x_encoder, mask, w1, w2, v
// ---------------------------------------------------------------------------
extern "C" void kernel_launch(void* const* d_in, const int* in_sizes, int n_in,
                              void* d_out, int out_size, void* d_ws, size_t ws_size,
                              hipStream_t stream) {
    const float* x_decoder = (const float*)d_in[0];
    const float* x_encoder = (const float*)d_in[1];
    const int*   mask      = (const int*)d_in[2];
    const float* w1        = (const float*)d_in[3];
    const float* w2        = (const float*)d_in[4];
    const float* v         = (const float*)d_in[5];
    float* out = (float*)d_out;

    float* enc_ws = (float*)d_ws;                       // 2 MB: (B*Nd, C)
    float* dec_ws = enc_ws + (size_t)BB * ND * CC;      // 2 MB: (B*Ne, C)

    dim3 gA(128, 16);   // (2048/16 M tiles) x (256/16 N tiles)
    proj_wmma_f32<<<gA, 32, 0, stream>>>(x_encoder, w1, enc_ws);
    proj_wmma_f32<<<gA, 32, 0, stream>>>(x_decoder, w2, dec_ws);

    dim3 gB(NE / 16, BB);   // 32 e-tiles x 4 batches, 512 thr = 16 waves
    attn_tanh_softmax<<<gB, 512, 0, stream>>>(enc_ws, dec_ws, mask, v, out);
}